// JointScorer_41429254537310
// MI455X (gfx1250) — compile-verified
//
#include <hip/hip_runtime.h>

// ---------------------------------------------------------------------------
// JointScorer for MI455X (gfx1250, wave32).
// GEMMs use V_WMMA_F32_16X16X4_F32 (exact f32, matrix pipe).
// For C = X * W^T both WMMA operands load as two consecutive floats per lane:
//   A 16x4 f32 layout: lanes 0-15 hold (M=lane, K=k+0,k+1), lanes 16-31 hold
//   (M=lane-16, K=k+2,k+3).  B 4x16 layout is the mirror with N=lane&15.
// The K loop runs in 16-wide regions, software-pipelined one region deep;
// sched_group_barrier pins the shape: 20 batched VMEM reads (next region's
// fragments), then 16 back-to-back WMMAs (current region) per region.
// ---------------------------------------------------------------------------

typedef __attribute__((ext_vector_type(2))) float v2f;
typedef __attribute__((ext_vector_type(8))) float v8f;

#define WMMA_F32(a, b, c) \
    __builtin_amdgcn_wmma_f32_16x16x4_f32(false, (a), false, (b), (short)0, (c), false, false)

#if __has_builtin(__builtin_amdgcn_sched_group_barrier)
#define SCHED_GROUP(mask, n) __builtin_amdgcn_sched_group_barrier((mask), (n), 0)
#else
#define SCHED_GROUP(mask, n)
#endif

// C[r, c] = act( sum_k A[r*lda + k] * W[c*ldw + koff + k] + bias[c] )
// grid.x = Cols/64, grid.y = Rows/128, block = 256 (8 waves, 16x64 tile/wave)
// K must be a multiple of 16 and >= 32 (here: 512 or 256).
__global__ __launch_bounds__(256) void gemm_xwt_wmma(
    const float* __restrict__ A, int lda,
    const float* __restrict__ W, int ldw, int koff,
    const float* __restrict__ bias, int do_relu,
    float* __restrict__ C, int ldc, int K)
{
    const int lane  = threadIdx.x & 31;
    const int wave  = threadIdx.x >> 5;
    const int row0  = (blockIdx.y * 8 + wave) * 16;
    const int col0  = blockIdx.x * 64;
    const int l15   = lane & 15;
    const int khalf = (lane >> 4) << 1;          // 0 for lanes 0-15, 2 for 16-31

    const int aoff  = (row0 + l15) * lda + khalf;
    const int woff  = (col0 + l15) * ldw + koff + khalf;
    const int wstep = 16 * ldw;

    v8f c0 = {}, c1 = {}, c2 = {}, c3 = {};

    v2f aF[4], bF[4][4];     // current region fragments (k .. k+12)
    v2f aN[4], bN[4][4];     // next region fragments    (k+16 .. k+28)

    // prologue: fragments for k = 0..12
#pragma unroll
    for (int s = 0; s < 4; ++s) {
        const int kk = 4 * s;
        aF[s]    = *(const v2f*)(A + aoff + kk);
        bF[0][s] = *(const v2f*)(W + woff + kk);
        bF[1][s] = *(const v2f*)(W + woff + wstep + kk);
        bF[2][s] = *(const v2f*)(W + woff + 2 * wstep + kk);
        bF[3][s] = *(const v2f*)(W + woff + 3 * wstep + kk);
    }

    for (int k = 0; k + 32 <= K; k += 16) {
        // batched loads for the NEXT region
#pragma unroll
        for (int s = 0; s < 4; ++s) {
            const int kk = k + 16 + 4 * s;
            aN[s]    = *(const v2f*)(A + aoff + kk);
            bN[0][s] = *(const v2f*)(W + woff + kk);
            bN[1][s] = *(const v2f*)(W + woff + wstep + kk);
            bN[2][s] = *(const v2f*)(W + woff + 2 * wstep + kk);
            bN[3][s] = *(const v2f*)(W + woff + 3 * wstep + kk);
        }
        // 16 WMMAs on the CURRENT region (loaded one region ago)
#pragma unroll
        for (int s = 0; s < 4; ++s) {
            c0 = WMMA_F32(aF[s], bF[0][s], c0);
            c1 = WMMA_F32(aF[s], bF[1][s], c1);
            c2 = WMMA_F32(aF[s], bF[2][s], c2);
            c3 = WMMA_F32(aF[s], bF[3][s], c3);
        }
        // rotate (resolved by register renaming)
#pragma unroll
        for (int s = 0; s < 4; ++s) {
            aF[s] = aN[s];
            bF[0][s] = bN[0][s];
            bF[1][s] = bN[1][s];
            bF[2][s] = bN[2][s];
            bF[3][s] = bN[3][s];
        }
        // pin the pipeline shape: 20 VMEM reads, then 16 WMMAs
        SCHED_GROUP(0x0020, 20);
        SCHED_GROUP(0x0008, 16);
    }

    // epilogue region: last 4 k-steps, no look-ahead loads
#pragma unroll
    for (int s = 0; s < 4; ++s) {
        c0 = WMMA_F32(aF[s], bF[0][s], c0);
        c1 = WMMA_F32(aF[s], bF[1][s], c1);
        c2 = WMMA_F32(aF[s], bF[2][s], c2);
        c3 = WMMA_F32(aF[s], bF[3][s], c3);
    }

    // C/D 16x16 f32 layout: VGPR v -> row (v + 8*(lane>>4)), col = lane&15
    const int mbase = (lane >> 4) * 8;
    v8f acc[4] = { c0, c1, c2, c3 };
#pragma unroll
    for (int j = 0; j < 4; ++j) {
        const int col = col0 + j * 16 + l15;
        const float bv = bias ? bias[col] : 0.0f;
#pragma unroll
        for (int v = 0; v < 8; ++v) {
            float val = acc[j][v] + bv;
            if (do_relu) val = fmaxf(val, 0.0f);
            C[(size_t)(row0 + mbase + v) * ldc + col] = val;
        }
    }
}

// t_emb[m, d] = E_pred[t0[m], d] + E_filt[t1[m], d] + E_reo[t2[m], d]
__global__ void embed_kernel(const int* __restrict__ tuples,
                             const float* __restrict__ Ep,
                             const float* __restrict__ Ef,
                             const float* __restrict__ Er,
                             float* __restrict__ t_emb, int M, int D)
{
    int idx = blockIdx.x * blockDim.x + threadIdx.x;
    if (idx >= M * D) return;
    int m = idx / D;
    int d = idx - m * D;
    int i0 = tuples[m * 3 + 0];
    int i1 = tuples[m * 3 + 1];
    int i2 = tuples[m * 3 + 2];
    t_emb[idx] = Ep[(size_t)i0 * D + d] + Ef[(size_t)i1 * D + d] + Er[(size_t)i2 * D + d];
}

// out[n, m] = sum_h relu(hx[n,h] + htb[m,h]) * w2[h] + b2[0]
// block = 256 (16x16 threads), 64x64 output tile, 4x4 outputs per thread.
#define TN 64
#define TM 64
#define HC 64
__global__ __launch_bounds__(256) void pairwise_kernel(
    const float* __restrict__ hx,   // [N, H]
    const float* __restrict__ htb,  // [M, H]  (b1 pre-folded)
    const float* __restrict__ w2,   // [H]
    const float* __restrict__ b2,   // [1]
    float* __restrict__ out, int N, int M, int H)
{
    __shared__ __align__(16) float sx[HC][TN + 4];  // transposed, stride 68 -> conflict-free b128
    __shared__ __align__(16) float st[HC][TM + 4];
    __shared__ float sw[HC];

    const int tid = threadIdx.x;
    const int tx = tid & 15;
    const int ty = tid >> 4;
    const int n0 = blockIdx.x * TN;
    const int m0 = blockIdx.y * TM;

    float acc[4][4] = {};

    for (int hc = 0; hc < H; hc += HC) {
        // cooperative staged loads: contiguous float4 along h, transpose into LDS
#pragma unroll
        for (int i = tid; i < TN * (HC / 4); i += 256) {
            int r  = i >> 4;          // HC/4 == 16
            int c4 = i & 15;
            float4 v = *(const float4*)(hx + (size_t)(n0 + r) * H + hc + c4 * 4);
            sx[c4 * 4 + 0][r] = v.x;
            sx[c4 * 4 + 1][r] = v.y;
            sx[c4 * 4 + 2][r] = v.z;
            sx[c4 * 4 + 3][r] = v.w;
        }
#pragma unroll
        for (int i = tid; i < TM * (HC / 4); i += 256) {
            int r  = i >> 4;
            int c4 = i & 15;
            float4 v = *(const float4*)(htb + (size_t)(m0 + r) * H + hc + c4 * 4);
            st[c4 * 4 + 0][r] = v.x;
            st[c4 * 4 + 1][r] = v.y;
            st[c4 * 4 + 2][r] = v.z;
            st[c4 * 4 + 3][r] = v.w;
        }
        if (tid < HC) sw[tid] = w2[hc + tid];
        __syncthreads();

        // prefetch next chunk's lines into cache while we grind VALU on this one
        if (hc + HC < H) {
            __builtin_prefetch(hx  + (size_t)(n0 + ty * 4) * H + hc + HC + tx * 4, 0, 0);
            __builtin_prefetch(htb + (size_t)(m0 + ty * 4) * H + hc + HC + tx * 4, 0, 0);
        }

#pragma unroll 4
        for (int h = 0; h < HC; ++h) {
            const float wv = sw[h];
            float4 xs = *(const float4*)(&sx[h][4 * ty]);
            float4 ts = *(const float4*)(&st[h][4 * tx]);
            const float xv[4] = { xs.x, xs.y, xs.z, xs.w };
            const float tv[4] = { ts.x, ts.y, ts.z, ts.w };
#pragma unroll
            for (int i = 0; i < 4; ++i)
#pragma unroll
                for (int j = 0; j < 4; ++j)
                    acc[i][j] = fmaf(fmaxf(xv[i] + tv[j], 0.0f), wv, acc[i][j]);
        }
        __syncthreads();
    }

    const float bb = b2[0];
#pragma unroll
    for (int i = 0; i < 4; ++i)
#pragma unroll
        for (int j = 0; j < 4; ++j)
            out[(size_t)(n0 + ty * 4 + i) * M + (m0 + tx * 4 + j)] = acc[i][j] + bb;
}

extern "C" void kernel_launch(void* const* d_in, const int* in_sizes, int n_in,
                              void* d_out, int out_size, void* d_ws, size_t ws_size,
                              hipStream_t stream)
{
    (void)in_sizes; (void)n_in; (void)out_size; (void)ws_size;

    const float* x      = (const float*)d_in[0];
    const int*   tuples = (const int*)  d_in[1];
    const float* Wa     = (const float*)d_in[2];
    const float* ba     = (const float*)d_in[3];
    const float* Wb     = (const float*)d_in[4];
    const float* bb     = (const float*)d_in[5];
    const float* Ep     = (const float*)d_in[6];
    const float* Ef     = (const float*)d_in[7];
    const float* Er     = (const float*)d_in[8];
    const float* W1     = (const float*)d_in[9];
    const float* b1     = (const float*)d_in[10];
    const float* W2     = (const float*)d_in[11];
    const float* b2     = (const float*)d_in[12];

    constexpr int N = 2048, M = 256, IN = 512, D = 256, H = 512;

    float* h_x   = (float*)d_ws;                 // [N, H]   4 MB
    float* x_emb = h_x   + (size_t)N * H;        // [N, D]   2 MB
    float* t_emb = x_emb + (size_t)N * D;        // [M, D]   256 KB
    float* hx    = t_emb + (size_t)M * D;        // [N, H]   4 MB
    float* htb   = hx    + (size_t)N * H;        // [M, H]   512 KB

    // h_x = relu(x @ Wa^T + ba)
    gemm_xwt_wmma<<<dim3(H / 64, N / 128), 256, 0, stream>>>(
        x, IN, Wa, IN, 0, ba, 1, h_x, H, IN);
    // x_emb = relu(h_x @ Wb^T + bb)
    gemm_xwt_wmma<<<dim3(D / 64, N / 128), 256, 0, stream>>>(
        h_x, H, Wb, H, 0, bb, 1, x_emb, D, H);
    // t_emb = Ep[t0] + Ef[t1] + Er[t2]
    embed_kernel<<<dim3((M * D + 255) / 256), 256, 0, stream>>>(
        tuples, Ep, Ef, Er, t_emb, M, D);
    // hx = x_emb @ W1[:, :D]^T
    gemm_xwt_wmma<<<dim3(H / 64, N / 128), 256, 0, stream>>>(
        x_emb, D, W1, 2 * D, 0, nullptr, 0, hx, H, D);
    // htb = t_emb @ W1[:, D:]^T + b1   (bias folded here)
    gemm_xwt_wmma<<<dim3(H / 64, M / 128), 256, 0, stream>>>(
        t_emb, D, W1, 2 * D, D, b1, 0, htb, H, D);
    // out = sum_h relu(hx + htb) * w2 + b2
    pairwise_kernel<<<dim3(N / 64, M / 64), 256, 0, stream>>>(
        hx, htb, W2, b2, (float*)d_out, N, M, H);
}